// RelationGraphSAGENetwork_14216341749899
// MI455X (gfx1250) — compile-verified
//
#include <hip/hip_runtime.h>
#include <hip/hip_bf16.h>

typedef float v2f __attribute__((ext_vector_type(2)));
typedef float v8f __attribute__((ext_vector_type(8)));

// ---------------------------------------------------------------------------
// Exact-fp32 WMMA: D(16x16) = A(16x4) * B(4x16) + C  -> v_wmma_f32_16x16x4_f32
// Lane layout (wave32): A/B frag = float2 per lane; lanes 0-15 carry K={0,1},
// lanes 16-31 carry K={2,3}. C/D: VGPR v holds row (v + 8*(lane>=16)), col lane%16.
// ---------------------------------------------------------------------------
__device__ inline v8f wmma4(v2f a, v2f b, v8f c) {
    return __builtin_amdgcn_wmma_f32_16x16x4_f32(
        /*neg_a=*/false, a, /*neg_b=*/false, b,
        /*c_mod=*/(short)0, c, /*reuse_a=*/false, /*reuse_b=*/false);
}

// ---------------------------------------------------------------------------
// Zero fill
// ---------------------------------------------------------------------------
__global__ void zero_kernel(float* __restrict__ p, long long n) {
    long long i = (long long)blockIdx.x * blockDim.x + threadIdx.x;
    if (i < n) p[i] = 0.0f;
}

// ---------------------------------------------------------------------------
// Per-destination degree and per-(dst, relation) counts (as floats, for GEMM)
// ---------------------------------------------------------------------------
__global__ void deg_cnt_kernel(const int* __restrict__ dst,
                               const int* __restrict__ rel,
                               float* __restrict__ deg,   // [N]
                               float* __restrict__ cnt,   // [N,16]
                               int E) {
    int e = blockIdx.x * blockDim.x + threadIdx.x;
    if (e >= E) return;
    int d = dst[e];
    atomicAdd(&deg[d], 1.0f);
    atomicAdd(&cnt[(long long)d * 16 + rel[e]], 1.0f);
}

// ---------------------------------------------------------------------------
// rel_msg[r][j] = sum_k rel_emb[r][k] * msg_W[j][128+k] + msg_b[j]   (16x128)
// ---------------------------------------------------------------------------
__global__ void rel_msg_kernel(const float* __restrict__ rel_emb_l,  // [16,32]
                               const float* __restrict__ MW,         // [128,160]
                               const float* __restrict__ MB,         // [128]
                               float* __restrict__ RMSG) {           // [16,128]
    int t = blockIdx.x * blockDim.x + threadIdx.x;
    if (t >= 16 * 128) return;
    int r = t >> 7, j = t & 127;
    const float* e = rel_emb_l + r * 32;
    const float* w = MW + (long long)j * 160 + 128;
    float acc = MB[j];
#pragma unroll
    for (int k = 0; k < 32; ++k) acc += e[k] * w[k];
    RMSG[t] = acc;
}

// ---------------------------------------------------------------------------
// S[dst] += H[src]  — one wave32 per edge, float4 per lane (512B/edge),
// global_atomic_add_f32 into the L2-resident accumulator.
// ---------------------------------------------------------------------------
__global__ __launch_bounds__(256) void scatter_kernel(
    const float* __restrict__ H, const int* __restrict__ src,
    const int* __restrict__ dst, float* __restrict__ S, int E) {
    int e = blockIdx.x * (blockDim.x >> 5) + (threadIdx.x >> 5);
    if (e >= E) return;
    int lane = threadIdx.x & 31;
    int s = src[e], d = dst[e];
    const float4 v = *(const float4*)(H + (long long)s * 128 + lane * 4);
    float* o = S + (long long)d * 128 + lane * 4;
    atomicAdd(o + 0, v.x);
    atomicAdd(o + 1, v.y);
    atomicAdd(o + 2, v.z);
    atomicAdd(o + 3, v.w);
}

// ---------------------------------------------------------------------------
// Y[N,128] = X[N,128] @ W[128,128]^T + b   (input projection), WMMA fp32
// Block = 256 thr = 8 waves; block handles 16 rows, wave w handles cols 16w..
// ---------------------------------------------------------------------------
__global__ __launch_bounds__(256) void gemm128_bias_kernel(
    const float* __restrict__ X, const float* __restrict__ W,
    const float* __restrict__ B, float* __restrict__ Y, int N) {
    const int wave = threadIdx.x >> 5;
    const int lane = threadIdx.x & 31;
    const int row0 = blockIdx.x * 16;
    const int n0   = wave * 16;
    const int mrow = lane & 15;
    const int hi   = lane >> 4;
    const int khi  = hi * 2;

    int ra = row0 + mrow; if (ra >= N) ra = N - 1;          // clamp tail reads
    const float* xrow = X + (long long)ra * 128;
    const float* wrow = W + (long long)(n0 + mrow) * 128;

    v8f c = {};
#pragma unroll
    for (int kk = 0; kk < 128; kk += 4) {
        v2f a  = *(const v2f*)(xrow + kk + khi);
        v2f bb = *(const v2f*)(wrow + kk + khi);
        c = wmma4(a, bb, c);
    }
    const int col  = n0 + mrow;
    const float bv = B[col];
    const int rb   = row0 + hi * 8;
#pragma unroll
    for (int v = 0; v < 8; ++v) {
        int r = rb + v;
        if (r < N) Y[(long long)r * 128 + col] = c[v] + bv;
    }
}

// ---------------------------------------------------------------------------
// Fused layer update:
//   agg = (S @ Wh^T + CNT @ RMSG) / max(deg,1)          (phase 1, via LDS)
//   Hout = relu(Hin @ SW^T + agg @ NW^T + SB + NB)      (phase 2)
// ---------------------------------------------------------------------------
__global__ __launch_bounds__(256) void layer_update_kernel(
    const float* __restrict__ Hin,  const float* __restrict__ S,
    const float* __restrict__ CNT,  const float* __restrict__ DEG,
    const float* __restrict__ RMSG, const float* __restrict__ MW,   // [128,160]
    const float* __restrict__ SW,   const float* __restrict__ SB,
    const float* __restrict__ NW,   const float* __restrict__ NB,
    float* __restrict__ Hout, int N) {
    __shared__ float agg[16 * 132];          // stride 132 -> conflict-free halves

    const int wave = threadIdx.x >> 5;
    const int lane = threadIdx.x & 31;
    const int row0 = blockIdx.x * 16;
    const int n0   = wave * 16;
    const int mrow = lane & 15;
    const int hi   = lane >> 4;
    const int khi  = hi * 2;

    int ra = row0 + mrow; if (ra >= N) ra = N - 1;

    // -------- phase 1: aggregate tile --------
    {
        const float* srow = S  + (long long)ra * 128;
        const float* wrow = MW + (long long)(n0 + mrow) * 160;  // Wh = cols 0..127
        v8f c = {};
#pragma unroll
        for (int kk = 0; kk < 128; kk += 4) {
            v2f a  = *(const v2f*)(srow + kk + khi);
            v2f bb = *(const v2f*)(wrow + kk + khi);
            c = wmma4(a, bb, c);
        }
        // relation-count contribution: A = CNT tile (16x16), B = RMSG (16x128)
        const float* crow = CNT + (long long)ra * 16;
#pragma unroll
        for (int kk = 0; kk < 16; kk += 4) {
            v2f a = *(const v2f*)(crow + kk + khi);
            v2f bb;
            bb.x = RMSG[(kk + khi + 0) * 128 + n0 + mrow];
            bb.y = RMSG[(kk + khi + 1) * 128 + n0 + mrow];
            c = wmma4(a, bb, c);
        }
        const int rb = hi * 8;
#pragma unroll
        for (int v = 0; v < 8; ++v) {
            int r = row0 + rb + v; if (r >= N) r = N - 1;
            float d = DEG[r];
            d = d < 1.0f ? 1.0f : d;
            agg[(rb + v) * 132 + n0 + mrow] = c[v] / d;
        }
    }
    __syncthreads();

    // -------- phase 2: node update --------
    {
        const float* hrow  = Hin + (long long)ra * 128;
        const float* swrow = SW  + (long long)(n0 + mrow) * 128;
        const float* nwrow = NW  + (long long)(n0 + mrow) * 128;
        const float* arow  = agg + mrow * 132;
        v8f c = {};
#pragma unroll
        for (int kk = 0; kk < 128; kk += 4) {
            v2f a1 = *(const v2f*)(hrow + kk + khi);
            v2f b1 = *(const v2f*)(swrow + kk + khi);
            c = wmma4(a1, b1, c);
            v2f a2 = *(const v2f*)(arow + kk + khi);   // LDS float2 (ds_load_b64)
            v2f b2 = *(const v2f*)(nwrow + kk + khi);
            c = wmma4(a2, b2, c);
        }
        const int col  = n0 + mrow;
        const float bv = SB[col] + NB[col];
        const int rb   = row0 + hi * 8;
#pragma unroll
        for (int v = 0; v < 8; ++v) {
            int r = rb + v;
            if (r < N) {
                float y = c[v] + bv;
                Hout[(long long)r * 128 + col] = y > 0.0f ? y : 0.0f;
            }
        }
    }
}

// ---------------------------------------------------------------------------
// Driver
// ---------------------------------------------------------------------------
extern "C" void kernel_launch(void* const* d_in, const int* in_sizes, int n_in,
                              void* d_out, int out_size, void* d_ws, size_t ws_size,
                              hipStream_t stream) {
    const float* x       = (const float*)d_in[0];
    const float* in_W    = (const float*)d_in[1];
    const float* in_b    = (const float*)d_in[2];
    const float* rel_emb = (const float*)d_in[3];   // [L,16,32]
    const float* msg_W   = (const float*)d_in[4];   // [L,128,160]
    const float* msg_b   = (const float*)d_in[5];   // [L,128]
    const float* self_W  = (const float*)d_in[6];   // [L,128,128]
    const float* self_b  = (const float*)d_in[7];   // [L,128]
    const float* neigh_W = (const float*)d_in[8];   // [L,128,128]
    const float* neigh_b = (const float*)d_in[9];   // [L,128]
    const int* edge_src  = (const int*)d_in[10];
    const int* edge_dst  = (const int*)d_in[11];
    const int* rel_ids   = (const int*)d_in[12];

    const int N = in_sizes[0] / 128;
    const int E = in_sizes[10];
    const int L = 2;
    float* out = (float*)d_out;

    // workspace layout
    float* h0   = (float*)d_ws;
    float* h1   = h0  + (long long)N * 128;
    float* Sbuf = h1  + (long long)N * 128;
    float* deg  = Sbuf + (long long)N * 128;
    float* cnt  = deg + N;                 // [N,16], contiguous with deg
    float* rmsg = cnt + (long long)N * 16; // [16,128]

    const int rowBlocks = (N + 15) / 16;

    // deg + cnt zero and accumulate (once; edges are layer-invariant)
    {
        long long nz = (long long)N * 17;
        zero_kernel<<<(unsigned)((nz + 255) / 256), 256, 0, stream>>>(deg, nz);
        deg_cnt_kernel<<<(E + 255) / 256, 256, 0, stream>>>(edge_dst, rel_ids, deg, cnt, E);
    }

    // input projection: h0 = x @ in_W^T + in_b
    gemm128_bias_kernel<<<rowBlocks, 256, 0, stream>>>(x, in_W, in_b, h0, N);

    for (int l = 0; l < L; ++l) {
        const float* hin  = (l == 0) ? h0 : h1;
        float*       hout = (l == L - 1) ? out : h1;

        // rel_msg table for this layer
        rel_msg_kernel<<<(16 * 128 + 255) / 256, 256, 0, stream>>>(
            rel_emb + (long long)l * 16 * 32,
            msg_W   + (long long)l * 128 * 160,
            msg_b   + (long long)l * 128, rmsg);

        // S = 0 ; S[dst] += h[src]
        {
            long long nz = (long long)N * 128;
            zero_kernel<<<(unsigned)((nz + 255) / 256), 256, 0, stream>>>(Sbuf, nz);
        }
        scatter_kernel<<<(E + 7) / 8, 256, 0, stream>>>(hin, edge_src, edge_dst, Sbuf, E);

        // fused aggregate + node update
        layer_update_kernel<<<rowBlocks, 256, 0, stream>>>(
            hin, Sbuf, cnt, deg, rmsg,
            msg_W   + (long long)l * 128 * 160,
            self_W  + (long long)l * 128 * 128,
            self_b  + (long long)l * 128,
            neigh_W + (long long)l * 128 * 128,
            neigh_b + (long long)l * 128,
            hout, N);
    }
}